// Vertical_space_layer_85272280695304
// MI455X (gfx1250) — compile-verified
//
#include <hip/hip_runtime.h>
#include <hip/hip_bf16.h>
#include <math.h>

typedef __attribute__((ext_vector_type(2))) float v2f;
typedef __attribute__((ext_vector_type(8))) float v8f;

#define B_ROWS 16384
#define D_DIM  1024
#define O_DIM  64
#define C_RAW  12
#define C_PAD  16
#define N_COLS (O_DIM * C_PAD)   // 1024 padded GEMM columns
#define KB     64                // K chunk staged in LDS

// ---------------------------------------------------------------------------
// Kernel 1: transpose + pad means into K-pair-major float2 layout.
// bt2[kp * N_COLS + n] = (means[o][c][2kp], means[o][c][2kp+1]) for n = o*16+c,
// zeros for padded c in [12,16).  One b64 load then gives a whole B fragment.
// ---------------------------------------------------------------------------
__global__ void k_prep_bt(const float* __restrict__ means, float2* __restrict__ bt2) {
  int idx = blockIdx.x * blockDim.x + threadIdx.x;   // 0 .. (D/2)*N_COLS - 1
  int n  = idx & (N_COLS - 1);
  int kp = idx >> 10;
  int o = n >> 4, c = n & 15;
  float2 v = make_float2(0.f, 0.f);
  if (c < C_RAW) {
    const float* s = means + ((size_t)(o * C_RAW + c) * D_DIM + (kp << 1));
    v.x = s[0]; v.y = s[1];
  }
  bt2[idx] = v;
}

// ---------------------------------------------------------------------------
// Kernel 2: x2[b] = sum_d x[b][d]^2.  One wave32 per row.
// ---------------------------------------------------------------------------
__global__ void k_x2(const float* __restrict__ x, float* __restrict__ x2out) {
  int row  = blockIdx.x * (blockDim.x >> 5) + (threadIdx.x >> 5);
  int lane = threadIdx.x & 31;
  const float* xr = x + (size_t)row * D_DIM;
  float s = 0.f;
  for (int i = lane; i < D_DIM; i += 32) { float v = xr[i]; s += v * v; }
  for (int m = 16; m; m >>= 1) s += __shfl_xor(s, m, 32);
  if (lane == 0) x2out[row] = s;
}

// ---------------------------------------------------------------------------
// Kernel 3: per padded column n = o*16+c: m2 = ||means[o,c]||^2 and
// inv2s2 = 1/(2*softplus(rho)^2); zeros for padded c. One wave32 per column.
// ---------------------------------------------------------------------------
__global__ void k_stats(const float* __restrict__ means, const float* __restrict__ rho,
                        float* __restrict__ m2, float* __restrict__ inv2s2) {
  int wid  = blockIdx.x * (blockDim.x >> 5) + (threadIdx.x >> 5); // 0..N_COLS-1
  int lane = threadIdx.x & 31;
  int o = wid >> 4, c = wid & 15;
  float s = 0.f, inv = 0.f;
  if (c < C_RAW) {
    const float* mr = means + (size_t)(o * C_RAW + c) * D_DIM;
    for (int i = lane; i < D_DIM; i += 32) { float v = mr[i]; s += v * v; }
    for (int m = 16; m; m >>= 1) s += __shfl_xor(s, m, 32);
    float r   = rho[o * C_RAW + c];
    float sig = log1pf(expf(r));               // softplus
    inv = 1.0f / (2.0f * sig * sig);
  }
  if (lane == 0) { m2[wid] = s; inv2s2[wid] = inv; }
}

// ---------------------------------------------------------------------------
// Main kernel: one workgroup (16 waves, 512 threads) per 16-row M tile.
// Wave w computes o-tiles w*4 .. w*4+3 with v_wmma_f32_16x16x4_f32,
// then applies the Gaussian-mixture epilogue via 16-lane shuffles.
// ---------------------------------------------------------------------------
__global__ __launch_bounds__(512, 1)
void k_main(const float* __restrict__ x, const float2* __restrict__ bt2,
            const float* __restrict__ x2arr, const float* __restrict__ m2arr,
            const float* __restrict__ invarr, float* __restrict__ out) {
  __shared__ float xs[16][68];                 // 16 x KB tile, pad 68 -> conflict-free A reads

  const int mtile = blockIdx.x;                // 0..1023
  const int tid   = threadIdx.x;
  const int wave  = tid >> 5;                  // 0..15
  const int lane  = tid & 31;
  const int half  = lane >> 4;                 // 0|1 (K sub-pair select)
  const int lc    = lane & 15;                 // column / row-in-tile index

  const float* xblk = x + (size_t)mtile * 16 * D_DIM;

  v8f acc[4] = {v8f{0.f}, v8f{0.f}, v8f{0.f}, v8f{0.f}};

  for (int kb = 0; kb < D_DIM; kb += KB) {
    // Cooperative LDS stage: 16 rows x 64 floats, 512 threads x 1 float2 each.
    {
      int r  = tid >> 5;                       // 0..15
      int cp = tid & 31;                       // float2 column 0..31
      float2 v = *((const float2*)(xblk + (size_t)r * D_DIM + kb) + cp);
      xs[r][cp * 2]     = v.x;
      xs[r][cp * 2 + 1] = v.y;
      if (kb + KB < D_DIM)                     // hint next chunk toward the caches
        __builtin_prefetch(xblk + (size_t)r * D_DIM + kb + KB + cp * 2, 0, 0);
    }
    __syncthreads();

    #pragma unroll
    for (int k0 = 0; k0 < KB; k0 += 4) {
      // A fragment (16x4 f32): lane row = lc, K = k0 + 2*half + {0,1} -> one ds b64
      v2f a = *(const v2f*)&xs[lc][k0 + 2 * half];
      #pragma unroll
      for (int t = 0; t < 4; ++t) {
        int o = wave * 4 + t;
        int n = o * 16 + lc;
        // B fragment (4x16 f32): K pair (kb+k0)/2 + half, column n -> one global b64
        float2 bv = bt2[(size_t)(((kb + k0) >> 1) + half) * N_COLS + n];
        v2f b; b.x = bv.x; b.y = bv.y;
        acc[t] = __builtin_amdgcn_wmma_f32_16x16x4_f32(
            false, a, false, b, (short)0, acc[t], false, false);
      }
    }
    __syncthreads();
  }

  // ---- epilogue ----
  // C/D layout: VGPR j, lane l -> row = j + 8*(l/16), col c = l%16.
  float x2r[8];
  #pragma unroll
  for (int j = 0; j < 8; ++j)
    x2r[j] = x2arr[mtile * 16 + 8 * half + j];

  const bool isG = (lc < 8);
  const bool isN = (lc >= 8) && (lc < C_RAW);

  #pragma unroll
  for (int t = 0; t < 4; ++t) {
    int o = wave * 4 + t;
    int n = o * 16 + lc;
    float m2v  = m2arr[n];
    float invv = invarr[n];
    #pragma unroll
    for (int j = 0; j < 8; ++j) {
      float xmv  = acc[t][j];
      float dist = x2r[j] + m2v - 2.0f * xmv;
      float G    = expf(-dist * invv);
      float gmax = isG ? G : -INFINITY;
      float gsum = isG ? G : 0.f;
      float nsum = isN ? G : 0.f;
      #pragma unroll
      for (int m = 1; m < 16; m <<= 1) {       // reduce across the 16 c-lanes
        gmax = fmaxf(gmax, __shfl_xor(gmax, m, 32));
        gsum += __shfl_xor(gsum, m, 32);
        nsum += __shfl_xor(nsum, m, 32);
      }
      // P = (max*9 - sum)/8 ; neg_P = nsum/4
      float result = (gmax * 9.0f - gsum) * 0.125f - nsum * 0.25f;
      if (lc == 0)
        out[(size_t)(mtile * 16 + 8 * half + j) * O_DIM + o] = result;
    }
  }
}

// ---------------------------------------------------------------------------
extern "C" void kernel_launch(void* const* d_in, const int* in_sizes, int n_in,
                              void* d_out, int out_size, void* d_ws, size_t ws_size,
                              hipStream_t stream) {
  const float* x     = (const float*)d_in[0];   // (16384, 1024)
  const float* means = (const float*)d_in[1];   // (64, 12, 1024)
  const float* rho   = (const float*)d_in[2];   // (64, 12)
  float* out = (float*)d_out;                   // (16384, 64)

  // workspace layout
  char* ws = (char*)d_ws;
  float2* bt2   = (float2*)ws;                                   // 4 MiB
  float*  x2a   = (float*)(ws + (size_t)(D_DIM / 2) * N_COLS * sizeof(float2));
  float*  m2a   = x2a + B_ROWS;
  float*  inva  = m2a + N_COLS;

  // 1) transpose/pad means: (D/2)*N_COLS elements, 256 threads/block
  k_prep_bt<<<(D_DIM / 2) * N_COLS / 256, 256, 0, stream>>>(means, bt2);
  // 2) x2: one wave per row, 8 waves per block
  k_x2<<<B_ROWS / 8, 256, 0, stream>>>(x, x2a);
  // 3) per-column stats: one wave per padded column
  k_stats<<<N_COLS / 8, 256, 0, stream>>>(means, rho, m2a, inva);
  // 4) main WMMA GEMM + epilogue: one workgroup per 16-row tile
  k_main<<<B_ROWS / 16, 512, 0, stream>>>(x, bt2, x2a, m2a, inva, out);
}